// AffineTransformer_12128987644032
// MI455X (gfx1250) — compile-verified
//
#include <hip/hip_runtime.h>
#include <hip/hip_bf16.h>
#include <stdint.h>

// Affine grid-sample (bilinear), N=16 C=32 H=512 W=512 -> 256x256, f32.
// Memory/latency bound (~0.6 GB traffic -> ~27us floor at 23.3 TB/s).
// One thread per (n,h,w); coords/weights computed once and amortized over
// all 32 channels. Per channel, the 4 bilinear taps are gathered with the
// CDNA5 async engine (global_load_async_to_lds_b32, ASYNCcnt) into a
// 4-deep LDS pipeline (3 stages / 12 loads in flight per wave), consumed
// as one ds_load_b128, and streamed out with non-temporal stores so L2
// stays reserved for the gather-side reuse.

#define N_MAXB    16
#define C_CONST   32
#define H_CONST   512
#define W_CONST   512
#define SH_CONST  256
#define SW_CONST  256
#define HW_CONST  (H_CONST * W_CONST)
#define OPLANE    (SH_CONST * SW_CONST)
#define NBUF      4
#define BUF_BYTES (256 * 4 * 4)   // 256 lanes * 4 taps * 4B = 4KB per buffer

__device__ __forceinline__ void async_tap_b32(unsigned lds_addr, const float* gaddr) {
    // CDNA5 async gather: memory -> LDS, per-lane address, tracked by ASYNCcnt.
    asm volatile("global_load_async_to_lds_b32 %0, %1, off"
                 :: "v"(lds_addr), "v"(gaddr)
                 : "memory");
}

__global__ __launch_bounds__(256)
void affine_bilinear_kernel(const float* __restrict__ x,
                            const float* __restrict__ params,
                            float* __restrict__ out) {
    __shared__ float stage[NBUF][256 * 4];   // 16KB: 4-deep tap pipeline

    const int tid = threadIdx.x;
    const int n   = blockIdx.y;
    const int pix = blockIdx.x * 256 + tid;      // 0 .. SH*SW-1 (exact cover)
    const int h   = pix / SW_CONST;
    const int w   = pix - h * SW_CONST;

    // Per-batch affine params (uniform within block -> scalarized loads).
    const float theta = params[n * 5 + 0];
    const float sx    = params[n * 5 + 1];
    const float sy    = params[n * 5 + 2];
    const float tx    = params[n * 5 + 3];
    const float ty    = params[n * 5 + 4];

    const float st = sinf(theta);
    const float ct = cosf(theta);
    const float Aa = sx * ct;   // A[0][0]
    const float Ab = -st;       // A[0][1]
    const float Ac = tx;        // A[0][2]
    const float Ad = st;        // A[1][0]
    const float Ae = sy * ct;   // A[1][1]
    const float Af = ty;        // A[1][2]

    // Source grid coords (mirror reference math exactly, fp32).
    const float wsf = floorf((float)w * ((float)W_CONST / (float)SW_CONST));
    const float hsf = floorf((float)h * ((float)H_CONST / (float)SH_CONST));
    const float gx  = (2.0f * wsf + 1.0f) / (float)W_CONST - 1.0f;
    const float gy  = (2.0f * hsf + 1.0f) / (float)H_CONST - 1.0f;

    const float sxg = Aa * gx + Ab * gy + Ac;
    const float syg = Ad * gx + Ae * gy + Af;
    const float ixf = ((sxg + 1.0f) * (float)W_CONST - 1.0f) * 0.5f;
    const float iyf = ((syg + 1.0f) * (float)H_CONST - 1.0f) * 0.5f;

    const float x0f = floorf(ixf), y0f = floorf(iyf);
    const float x1f = x0f + 1.0f,  y1f = y0f + 1.0f;
    const float fx1 = ixf - x0f,   fx0 = 1.0f - fx1;
    const float fy1 = iyf - y0f,   fy0 = 1.0f - fy1;

    const bool vx0 = (x0f >= 0.0f) && (x0f <= (float)(W_CONST - 1));
    const bool vx1 = (x1f >= 0.0f) && (x1f <= (float)(W_CONST - 1));
    const bool vy0 = (y0f >= 0.0f) && (y0f <= (float)(H_CONST - 1));
    const bool vy1 = (y1f >= 0.0f) && (y1f <= (float)(H_CONST - 1));

    // Fold validity into the tap weights (v * valid * w == v * (valid ? w : 0)).
    const float w00 = (vy0 && vx0) ? fy0 * fx0 : 0.0f;
    const float w01 = (vy0 && vx1) ? fy0 * fx1 : 0.0f;
    const float w10 = (vy1 && vx0) ? fy1 * fx0 : 0.0f;
    const float w11 = (vy1 && vx1) ? fy1 * fx1 : 0.0f;

    const int x0c = min(max((int)x0f, 0), W_CONST - 1);
    const int x1c = min(max((int)x1f, 0), W_CONST - 1);
    const int y0c = min(max((int)y0f, 0), H_CONST - 1);
    const int y1c = min(max((int)y1f, 0), H_CONST - 1);

    const float* xn  = x + (size_t)n * C_CONST * HW_CONST;
    const float* p00 = xn + (y0c * W_CONST + x0c);
    const float* p01 = xn + (y0c * W_CONST + x1c);
    const float* p10 = xn + (y1c * W_CONST + x0c);
    const float* p11 = xn + (y1c * W_CONST + x1c);

    // Allocation-relative LDS byte address of this lane's 16B tap slot
    // (flat addr low 32 bits == LDS offset).
    const unsigned lme = (unsigned)(uintptr_t)(&stage[0][tid * 4]);

    float* outn = out + (size_t)n * C_CONST * OPLANE + pix;

    // Stage channel c's 4 taps into LDS buffer (c & 3).
    auto stage_ch = [&](int c) {
        const size_t   coff = (size_t)c * HW_CONST;
        const unsigned l    = lme + (unsigned)(c & (NBUF - 1)) * (unsigned)BUF_BYTES;
        async_tap_b32(l + 0,  p00 + coff);
        async_tap_b32(l + 4,  p01 + coff);
        async_tap_b32(l + 8,  p10 + coff);
        async_tap_b32(l + 12, p11 + coff);
    };

    // Consume channel c from LDS: one ds_load_b128, blend, NT store.
    auto consume_ch = [&](int c) {
        const float4 v = *(const float4*)(&stage[c & (NBUF - 1)][tid * 4]);
        const float  r = v.x * w00 + v.y * w01 + v.z * w10 + v.w * w11;
        __builtin_nontemporal_store(r, outn + (size_t)c * OPLANE);
    };

    // Prologue: fill 3 pipeline stages (12 async loads in flight).
    stage_ch(0);
    stage_ch(1);
    stage_ch(2);

    // Steady state: keep 3 stages (12 loads) in flight per wave.
    for (int c = 0; c < C_CONST - 3; ++c) {
        // WAR guard: buffer (c+3)&3 == (c-1)&3 was read by DS loads last
        // iteration; make sure those retired before the async engine
        // overwrites it.
        asm volatile("s_wait_dscnt 0x0" ::: "memory");
        stage_ch(c + 3);
        // In-order async completion: <=12 outstanding => stage c has fully
        // landed in LDS while stages c+1..c+3 remain in flight.
        asm volatile("s_wait_asynccnt 0xc" ::: "memory");
        consume_ch(c);
    }

    // Epilogue: drain the pipeline with constant-immediate waits.
    asm volatile("s_wait_asynccnt 0x8" ::: "memory");
    consume_ch(C_CONST - 3);
    asm volatile("s_wait_asynccnt 0x4" ::: "memory");
    consume_ch(C_CONST - 2);
    asm volatile("s_wait_asynccnt 0x0" ::: "memory");
    consume_ch(C_CONST - 1);
}

extern "C" void kernel_launch(void* const* d_in, const int* in_sizes, int n_in,
                              void* d_out, int out_size, void* d_ws, size_t ws_size,
                              hipStream_t stream) {
    const float* x      = (const float*)d_in[0];
    const float* params = (const float*)d_in[1];
    // d_in[2]/d_in[3] are small_h/small_w (=256,256 per setup_inputs); grid
    // geometry must be host-side, so we use the reference constants.
    float* out = (float*)d_out;

    int N = (n_in > 1) ? (in_sizes[1] / 5) : N_MAXB;
    if (N <= 0) N = N_MAXB;

    dim3 grid(OPLANE / 256, N, 1);
    dim3 block(256, 1, 1);
    affine_bilinear_kernel<<<grid, block, 0, stream>>>(x, params, out);
}